// EquivLayerNorm_20031727469174
// MI455X (gfx1250) — compile-verified
//
#include <hip/hip_runtime.h>
#include <stdint.h>

// EquivLayerNorm: irreps = 128x0e + 64x1o + 32x2e, DIM=480, rows=200000.
// Bandwidth-bound: 768 MB total traffic -> ~33us floor at 23.3 TB/s.
// Strategy: one wave32 per row, double-buffered async global->LDS staging
// (CDNA5 GLOBAL_LOAD_ASYNC_TO_LDS_B128 with NT hint + s_wait_asynccnt),
// shfl_xor wave reductions, non-temporal float4 stores (stream > L2 size).

#define EQ_DIM 480
#define EQ_EPS 1e-8f

typedef float v4f __attribute__((ext_vector_type(4)));

__device__ __forceinline__ float wave_sum32(float v) {
#pragma unroll
  for (int off = 16; off; off >>= 1) v += __shfl_xor(v, off, 32);
  return v;
}

// Copy one 1920-byte row from global memory into LDS asynchronously.
// 4 x b128 per wave (last one masked to lanes 0..23: 480 floats = 3.75 * 512B).
// INST_OFFSET is added to BOTH the global and the LDS address (ISA 08 §4.4).
// NT hint: rows are read exactly once; don't displace L2 contents.
__device__ __forceinline__ void async_row(const float* __restrict__ g,
                                          uint32_t lds_base, int lane) {
  uint64_t ga = (uint64_t)(uintptr_t)g + (uint64_t)(lane * 16);
  uint32_t la = lds_base + (uint32_t)(lane * 16);
  asm volatile("global_load_async_to_lds_b128 %0, %1, off th:TH_LOAD_NT"
               :: "v"(la), "v"(ga) : "memory");
  asm volatile("global_load_async_to_lds_b128 %0, %1, off offset:512 th:TH_LOAD_NT"
               :: "v"(la), "v"(ga) : "memory");
  asm volatile("global_load_async_to_lds_b128 %0, %1, off offset:1024 th:TH_LOAD_NT"
               :: "v"(la), "v"(ga) : "memory");
  if (lane < 24) {
    asm volatile("global_load_async_to_lds_b128 %0, %1, off offset:1536 th:TH_LOAD_NT"
                 :: "v"(la), "v"(ga) : "memory");
  }
}

__global__ __launch_bounds__(256) void eqln_kernel(
    const float* __restrict__ feat, const float* __restrict__ scale,
    const float* __restrict__ offset0, float* __restrict__ out, int nrows) {
  // 8 waves per block, each wave owns a private 2 x 480-float double buffer.
  __shared__ float lbuf[8 * 2 * EQ_DIM];

  const int lane = threadIdx.x & 31;
  const int wid  = threadIdx.x >> 5;
  const int gw   = blockIdx.x * 8 + wid;          // global wave id
  const int nw   = gridDim.x * 8;                 // total waves

  const float s0 = scale[0], s1 = scale[1], s2 = scale[2];
  const float off0 = offset0[0];

  float* buf = &lbuf[wid * (2 * EQ_DIM)];
  // Low 32 bits of a generic shared-space address are the LDS byte offset.
  const uint32_t l0 = (uint32_t)(uintptr_t)buf;
  const uint32_t l1 = l0 + EQ_DIM * 4;

  int row = gw;
  if (row < nrows) async_row(feat + (size_t)row * EQ_DIM, l0, lane);
  int cur = 0;

  for (; row < nrows; row += nw) {
    const int nxt = row + nw;
    if (nxt < nrows) {
      // Prefetch next row into the other buffer, then wait until the 4
      // outstanding loads of the CURRENT buffer are done (in-order count).
      async_row(feat + (size_t)nxt * EQ_DIM, cur ? l0 : l1, lane);
      asm volatile("s_wait_asynccnt 4" ::: "memory");
    } else {
      asm volatile("s_wait_asynccnt 0" ::: "memory");
    }

    const float4* s4 = (const float4*)(buf + cur * EQ_DIM);
    float4 v0 = s4[lane];            // chunks   0.. 31 : 0e scalars
    float4 v1 = s4[lane + 32];       // chunks  32.. 63 : 1o
    float4 v2 = s4[lane + 64];       // chunks  64.. 95 : 1o (lane<16) / 2e
    float4 v3 = make_float4(0.f, 0.f, 0.f, 0.f);
    if (lane < 24) v3 = s4[lane + 96];  // chunks 96..119 : 2e

    // Mean over the 128 scalars; subtract only from scalars (mask behavior).
    const float m =
        wave_sum32(v0.x + v0.y + v0.z + v0.w) * (1.0f / 128.0f);
    v0.x -= m; v0.y -= m; v0.z -= m; v0.w -= m;

    const float q0 = v0.x * v0.x + v0.y * v0.y + v0.z * v0.z + v0.w * v0.w;
    const float q1 = v1.x * v1.x + v1.y * v1.y + v1.z * v1.z + v1.w * v1.w;
    const float q2 = v2.x * v2.x + v2.y * v2.y + v2.z * v2.z + v2.w * v2.w;
    const float q3 = v3.x * v3.x + v3.y * v3.y + v3.z * v3.z + v3.w * v3.w;
    const bool hi = (lane >= 16);

    const float ss0 = wave_sum32(q0);
    const float ss1 = wave_sum32(q1 + (hi ? 0.f : q2));
    const float ss2 = wave_sum32(q3 + (hi ? q2 : 0.f));

    // Unbiased divisors: 127 / 63 / 31, fold scale into the inverse norm.
    const float i0 = s0 / sqrtf(ss0 * (1.0f / 127.0f) + EQ_EPS);
    const float i1 = s1 / sqrtf(ss1 * (1.0f / 63.0f) + EQ_EPS);
    const float i2 = s2 / sqrtf(ss2 * (1.0f / 31.0f) + EQ_EPS);
    const float im = hi ? i2 : i1;

    // Non-temporal streaming stores: output written once, never re-read.
    v4f* o4 = (v4f*)(out + (size_t)row * EQ_DIM);
    v4f o0 = {v0.x * i0 + off0, v0.y * i0 + off0,
              v0.z * i0 + off0, v0.w * i0 + off0};
    v4f o1 = {v1.x * i1, v1.y * i1, v1.z * i1, v1.w * i1};
    v4f o2 = {v2.x * im, v2.y * im, v2.z * im, v2.w * im};
    __builtin_nontemporal_store(o0, o4 + lane);
    __builtin_nontemporal_store(o1, o4 + lane + 32);
    __builtin_nontemporal_store(o2, o4 + lane + 64);
    if (lane < 24) {
      v4f o3 = {v3.x * i2, v3.y * i2, v3.z * i2, v3.w * i2};
      __builtin_nontemporal_store(o3, o4 + lane + 96);
    }

    cur ^= 1;
  }
}

extern "C" void kernel_launch(void* const* d_in, const int* in_sizes, int n_in,
                              void* d_out, int out_size, void* d_ws,
                              size_t ws_size, hipStream_t stream) {
  const float* feat    = (const float*)d_in[0];
  const float* scale   = (const float*)d_in[1];
  const float* offset0 = (const float*)d_in[2];
  float* out           = (float*)d_out;

  const int nrows = in_sizes[0] / EQ_DIM;   // 200000
  int blocks = (nrows + 7) / 8;             // 8 waves (rows) per 256-thr block
  if (blocks > 2560) blocks = 2560;         // ~10 rows/wave for pipelining
  if (blocks < 1) blocks = 1;

  hipLaunchKernelGGL(eqln_kernel, dim3(blocks), dim3(256), 0, stream,
                     feat, scale, offset0, out, nrows);
}